// GCNModel_48112223650296
// MI455X (gfx1250) — compile-verified
//
#include <hip/hip_runtime.h>
#include <hip/hip_bf16.h>

// ---------------------------------------------------------------------------
// 3-layer GCN for MI455X (gfx1250, wave32).
//   Layer l:  H = Z @ Wl  (WMMA f32 16x16x4, weights staged pair-packed in LDS)
//             A[i] = dinv[i]^2 * H[i]                      (self-loop seed)
//             A[dst] += dinv[src]*dinv[dst]*H[src]         (edge scatter,
//                        inline-asm global_atomic_add_f32, device scope)
//             Z = relu(A + b)
//   Final 64->2 layer with plain FMA, then log_softmax.
// Memory-bound on edge gather/scatter (~2 GB @ 23.3 TB/s); GEMMs ~2.5 GFLOP.
// ---------------------------------------------------------------------------

typedef float v2f __attribute__((ext_vector_type(2)));
typedef float v8f __attribute__((ext_vector_type(8)));

#define N_NODES 100000

// Hardware fp32 atomic add at device (L2) scope — no CAS loop, no return.
__device__ __forceinline__ void atomAddF(float* p, float v) {
  asm volatile("global_atomic_add_f32 %0, %1, off scope:SCOPE_DEV"
               :: "v"(p), "v"(v) : "memory");
}

// ---------------- degree / norm ----------------
__global__ void deg_init(float* deg, int n) {
  int i = blockIdx.x * blockDim.x + threadIdx.x;
  if (i < n) deg[i] = 1.0f;              // self-loop contributes 1 to every degree
}

__global__ void deg_accum(const int* __restrict__ dst, float* deg, int e) {
  int i = blockIdx.x * blockDim.x + threadIdx.x;
  if (i < e) atomAddF(&deg[dst[i]], 1.0f);
}

__global__ void deg_to_dinv(float* deg, int n) {
  int i = blockIdx.x * blockDim.x + threadIdx.x;
  if (i < n) deg[i] = rsqrtf(deg[i]);    // deg >= 1 always
}

// ---------------- dense transform: H[M x 64] = X[M x K] * W[K x 64] ----------
// One wave -> one 16-row tile x all 64 cols (4 WMMA f32 16x16x4 accumulators).
// W staged in LDS pair-packed: lds[pr*128 + 2*c + {0,1}] = W[2*pr + {0,1}][c],
// so a B fragment {W[k][c], W[k+1][c]} is a single aligned ds_load_b64.
__global__ void __launch_bounds__(256)
gemm_wmma_f32(const float* __restrict__ X, const float* __restrict__ W,
              float* __restrict__ H, int M, int K) {
  extern __shared__ float lds[];
  const int tid = threadIdx.x;
  const int nPairs = (K >> 1) * 64;                    // K*32 pair elements
  for (int i = tid; i < nPairs; i += blockDim.x) {
    int pr = i >> 6;                                   // pair-row (rows 2pr, 2pr+1)
    int c  = i & 63;                                   // column
    v2f p = { W[(2 * pr) * 64 + c], W[(2 * pr + 1) * 64 + c] };
    *(v2f*)(lds + (size_t)pr * 128 + 2 * c) = p;       // coalesced read, b64 write
  }
  __syncthreads();

  const int wave = tid >> 5;
  const int lane = tid & 31;
  const int tile = blockIdx.x * 8 + wave;              // 16-row tile index
  if (tile * 16 >= M) return;                          // wave-uniform exit
  const int m0 = tile * 16;
  const int r  = lane & 15;                            // row/col within tile
  const int kh = lane >> 4;                            // K-half select

  v8f c0 = {}, c1 = {}, c2 = {}, c3 = {};
  const float* xrow = X + (size_t)(m0 + r) * K;
  for (int k = 0; k < K; k += 4) {
    // A 16x4: lanes 0-15 hold K={k,k+1}, lanes 16-31 hold K={k+2,k+3}
    v2f a = *(const v2f*)(xrow + k + 2 * kh);
    // pair-row for K rows (k+2kh, k+2kh+1); k+2kh is even
    const float* wb = lds + (size_t)((k + 2 * kh) >> 1) * 128;
    v2f b0 = *(const v2f*)(wb + 2 * ( 0 + r));
    v2f b1 = *(const v2f*)(wb + 2 * (16 + r));
    v2f b2 = *(const v2f*)(wb + 2 * (32 + r));
    v2f b3 = *(const v2f*)(wb + 2 * (48 + r));
    c0 = __builtin_amdgcn_wmma_f32_16x16x4_f32(false, a, false, b0, (short)0, c0, false, false);
    c1 = __builtin_amdgcn_wmma_f32_16x16x4_f32(false, a, false, b1, (short)0, c1, false, false);
    c2 = __builtin_amdgcn_wmma_f32_16x16x4_f32(false, a, false, b2, (short)0, c2, false, false);
    c3 = __builtin_amdgcn_wmma_f32_16x16x4_f32(false, a, false, b3, (short)0, c3, false, false);
  }
  // C/D layout: VGPR v, lanes 0-15 -> M=v, lanes 16-31 -> M=v+8, N=r
#pragma unroll
  for (int v = 0; v < 8; ++v) {
    float* hrow = H + (size_t)(m0 + v + 8 * kh) * 64;
    hrow[ 0 + r] = c0[v];
    hrow[16 + r] = c1[v];
    hrow[32 + r] = c2[v];
    hrow[48 + r] = c3[v];
  }
}

// ---------------- self-loop seed: A[i] = dinv[i]^2 * H[i]  (D = 64) ---------
__global__ void seed_self64(const float* __restrict__ dinv, const float* __restrict__ H,
                            float* __restrict__ A, int n) {
  int idx = blockIdx.x * blockDim.x + threadIdx.x;   // over n*16 float4 chunks
  if (idx >= n * 16) return;
  int i = idx >> 4, c = idx & 15;
  float w = dinv[i] * dinv[i];
  float4 h = ((const float4*)(H + (size_t)i * 64))[c];
  float4 o = { w * h.x, w * h.y, w * h.z, w * h.w };
  ((float4*)(A + (size_t)i * 64))[c] = o;
}

// ---------------- edge scatter: A[dst] += dinv[s]*dinv[d]*H[src] (D = 64) ---
// 16 consecutive lanes share one edge -> coalesced float4 gather + scatter.
__global__ void edge_scatter64(const int* __restrict__ src, const int* __restrict__ dst,
                               const float* __restrict__ dinv, const float* __restrict__ H,
                               float* __restrict__ A, int e) {
  int idx = blockIdx.x * blockDim.x + threadIdx.x;   // over e*16
  if (idx >= e * 16) return;
  int ed = idx >> 4, c = idx & 15;
  int s = src[ed], d = dst[ed];
  float w = dinv[s] * dinv[d];
  float4 h = ((const float4*)(H + (size_t)s * 64))[c];
  float* out = A + (size_t)d * 64 + c * 4;
  atomAddF(out + 0, w * h.x);
  atomAddF(out + 1, w * h.y);
  atomAddF(out + 2, w * h.z);
  atomAddF(out + 3, w * h.w);
}

// ---------------- Z = relu(A + b)  (D = 64) ---------------------------------
__global__ void bias_relu64(const float* __restrict__ A, const float* __restrict__ b,
                            float* __restrict__ Z, int n) {
  int idx = blockIdx.x * blockDim.x + threadIdx.x;   // over n*64
  if (idx >= n * 64) return;
  Z[idx] = fmaxf(A[idx] + b[idx & 63], 0.0f);
}

// ---------------- final tiny transform: H[N x 2] = Z[N x 64] * W3[64 x 2] ---
__global__ void gemm_out2(const float* __restrict__ Z, const float* __restrict__ W3,
                          float* __restrict__ H, int n) {
  int i = blockIdx.x * blockDim.x + threadIdx.x;
  if (i >= n) return;
  const float* z = Z + (size_t)i * 64;
  float a0 = 0.f, a1 = 0.f;
#pragma unroll
  for (int f = 0; f < 64; ++f) {
    float zf = z[f];
    a0 = fmaf(zf, W3[f * 2 + 0], a0);
    a1 = fmaf(zf, W3[f * 2 + 1], a1);
  }
  H[(size_t)i * 2 + 0] = a0;
  H[(size_t)i * 2 + 1] = a1;
}

__global__ void seed_self2(const float* __restrict__ dinv, const float* __restrict__ H,
                           float* __restrict__ A, int n) {
  int i = blockIdx.x * blockDim.x + threadIdx.x;
  if (i >= n) return;
  float w = dinv[i] * dinv[i];
  A[(size_t)i * 2 + 0] = w * H[(size_t)i * 2 + 0];
  A[(size_t)i * 2 + 1] = w * H[(size_t)i * 2 + 1];
}

__global__ void edge_scatter2(const int* __restrict__ src, const int* __restrict__ dst,
                              const float* __restrict__ dinv, const float* __restrict__ H,
                              float* __restrict__ A, int e) {
  int i = blockIdx.x * blockDim.x + threadIdx.x;
  if (i >= e) return;
  int s = src[i], d = dst[i];
  float w = dinv[s] * dinv[d];
  atomAddF(A + (size_t)d * 2 + 0, w * H[(size_t)s * 2 + 0]);
  atomAddF(A + (size_t)d * 2 + 1, w * H[(size_t)s * 2 + 1]);
}

__global__ void final_logsoftmax(const float* __restrict__ A, const float* __restrict__ b3,
                                 float* __restrict__ out, int n) {
  int i = blockIdx.x * blockDim.x + threadIdx.x;
  if (i >= n) return;
  float v0 = A[(size_t)i * 2 + 0] + b3[0];
  float v1 = A[(size_t)i * 2 + 1] + b3[1];
  float m  = fmaxf(v0, v1);
  float lse = m + logf(expf(v0 - m) + expf(v1 - m));
  out[(size_t)i * 2 + 0] = v0 - lse;
  out[(size_t)i * 2 + 1] = v1 - lse;
}

// ---------------------------------------------------------------------------
static inline int cdiv(long long a, int b) { return (int)((a + b - 1) / b); }

extern "C" void kernel_launch(void* const* d_in, const int* in_sizes, int n_in,
                              void* d_out, int out_size, void* d_ws, size_t ws_size,
                              hipStream_t stream) {
  const float* x   = (const float*)d_in[0];           // [N, 128]
  const int*   ei  = (const int*)d_in[1];             // [2, E]
  const float* W1  = (const float*)d_in[2];           // [128, 64]
  const float* b1  = (const float*)d_in[3];
  const float* W2  = (const float*)d_in[4];           // [64, 64]
  const float* b2  = (const float*)d_in[5];
  const float* W3  = (const float*)d_in[6];           // [64, 2]
  const float* b3  = (const float*)d_in[7];
  float* out = (float*)d_out;

  const int N = N_NODES;
  const int E = in_sizes[1] / 2;
  const int* src = ei;
  const int* dst = ei + E;

  // workspace: dinv[N] | H[N*64] | A[N*64] | Z[N*64]  (~77 MB of f32)
  float* dinv = (float*)d_ws;
  float* H = dinv + N;
  float* A = H + (size_t)N * 64;
  float* Z = A + (size_t)N * 64;

  const int B = 256;
  const int gemmGrid = cdiv((long long)(N / 16), 8);  // N multiple of 16; 8 waves/block

  // --- normalization ---
  deg_init   <<<cdiv(N, B), B, 0, stream>>>(dinv, N);
  deg_accum  <<<cdiv(E, B), B, 0, stream>>>(dst, dinv, E);
  deg_to_dinv<<<cdiv(N, B), B, 0, stream>>>(dinv, N);

  // --- layer 1: 128 -> 64 ---
  gemm_wmma_f32<<<gemmGrid, B, 128 * 64 * sizeof(float), stream>>>(x, W1, H, N, 128);
  seed_self64  <<<cdiv((long long)N * 16, B), B, 0, stream>>>(dinv, H, A, N);
  edge_scatter64<<<cdiv((long long)E * 16, B), B, 0, stream>>>(src, dst, dinv, H, A, E);
  bias_relu64  <<<cdiv((long long)N * 64, B), B, 0, stream>>>(A, b1, Z, N);

  // --- layer 2: 64 -> 64 ---
  gemm_wmma_f32<<<gemmGrid, B, 64 * 64 * sizeof(float), stream>>>(Z, W2, H, N, 64);
  seed_self64  <<<cdiv((long long)N * 16, B), B, 0, stream>>>(dinv, H, A, N);
  edge_scatter64<<<cdiv((long long)E * 16, B), B, 0, stream>>>(src, dst, dinv, H, A, E);
  bias_relu64  <<<cdiv((long long)N * 64, B), B, 0, stream>>>(A, b2, Z, N);

  // --- layer 3: 64 -> 2, then log_softmax ---
  gemm_out2    <<<cdiv(N, B), B, 0, stream>>>(Z, W3, H, N);
  seed_self2   <<<cdiv(N, B), B, 0, stream>>>(dinv, H, A, N);
  edge_scatter2<<<cdiv(E, B), B, 0, stream>>>(src, dst, dinv, H, A, E);
  final_logsoftmax<<<cdiv(N, B), B, 0, stream>>>(A, b3, out, N);
}